// _RoIPooling_77446850281575
// MI455X (gfx1250) — compile-verified
//
#include <hip/hip_runtime.h>
#include <cmath>

// RoI max pooling for MI455X (gfx1250, wave32).
// features: [4,256,64,64] f32, rois: [1024,5] f32, out: [1024,256,7,7] f32.

#define PH 7
#define PW 7
#define RCH 256      // channels
#define FH 64
#define FW 64
#define BINS (PH * PW)                 // 49
#define PER_ROI (RCH * BINS)           // 12544 outputs per ROI (multiple of 256)
#define BLOCKS_PER_ROI (PER_ROI / 256) // 49 blocks of 256 threads

__global__ __launch_bounds__(256) void _RoIPooling_gfx1250_kernel(
    const float* __restrict__ feat,
    const float* __restrict__ rois,
    float* __restrict__ out)
{
    __shared__ float s_roi[8];                 // 5 ROI floats (padded)
    __shared__ int   s_bidx;
    __shared__ int   s_hs[PH], s_he[PH], s_ws[PW], s_we[PW];

    const int tid = threadIdx.x;
    const int rb  = blockIdx.x;                // 0 .. R*49-1
    const int r   = rb / BLOCKS_PER_ROI;       // ROI index (block-uniform)
    const int j   = rb - r * BLOCKS_PER_ROI;   // which 256-output slice of this ROI

    // ---- Stage this ROI's 5 floats into LDS via CDNA5 async global->LDS DMA.
    // Generic address of a __shared__ var is {LDS aperture, lds_offset[31:0]};
    // low 32 bits give the wave-relative LDS byte address the instruction needs.
    if (tid < 5) {
        unsigned lds_off = (unsigned)(unsigned long long)(&s_roi[tid]);
        unsigned long long gaddr =
            (unsigned long long)(const void*)(rois + (size_t)r * 5 + tid);
        asm volatile("global_load_async_to_lds_b32 %0, %1, off"
                     :: "v"(lds_off), "v"(gaddr) : "memory");
    }
    asm volatile("s_wait_asynccnt 0" ::: "memory");
    __syncthreads();

    // ---- Per-block bin tables (7 threads; round-half-even matches jnp.round).
    if (tid < PH) {
        const float sc = 1.0f / 16.0f;
        float x1 = s_roi[1], y1 = s_roi[2], x2 = s_roi[3], y2 = s_roi[4];
        int rsw = (int)rintf(x1 * sc);
        int rsh = (int)rintf(y1 * sc);
        int rew = (int)rintf(x2 * sc);
        int reh = (int)rintf(y2 * sc);
        float roi_h = (float)max(reh - rsh + 1, 1);
        float roi_w = (float)max(rew - rsw + 1, 1);
        float bsh = roi_h / (float)PH;
        float bsw = roi_w / (float)PW;
        int p  = tid;
        int hs = (int)floorf((float)p * bsh) + rsh;
        int he = (int)ceilf(((float)p + 1.0f) * bsh) + rsh;
        int ws = (int)floorf((float)p * bsw) + rsw;
        int we = (int)ceilf(((float)p + 1.0f) * bsw) + rsw;
        s_hs[p] = min(max(hs, 0), FH);
        s_he[p] = min(max(he, 0), FH);
        s_ws[p] = min(max(ws, 0), FW);
        s_we[p] = min(max(we, 0), FW);
        if (p == 0) s_bidx = (int)s_roi[0];
    }
    __syncthreads();

    // ---- One output element per thread; consecutive tid -> consecutive
    // output index, so each wave32 issues one coalesced 128B store.
    const int lidx = j * 256 + tid;            // 0 .. 12543 within this ROI
    const int c    = lidx / BINS;
    const int phw  = lidx - c * BINS;
    const int ph   = phw / PW;
    const int pw   = phw - ph * PW;

    const int hs  = s_hs[ph], he = s_he[ph];
    const int wss = s_ws[pw], wee = s_we[pw];
    const float* base = feat + ((size_t)(s_bidx * RCH + c)) * (FH * FW);

    float m;
    if (hs >= he || wss >= wee) {
        m = 0.0f;                              // empty bin -> 0 (reference rule)
    } else {
        // CDNA5 prefetch of this thread's first feature row (global_prefetch_b8)
        __builtin_prefetch(base + hs * FW + wss, 0, 0);
        m = -__builtin_inff();
        for (int h = hs; h < he; ++h) {
            const float* row = base + h * FW;
            for (int w = wss; w < wee; ++w) {
                m = fmaxf(m, row[w]);
            }
        }
    }

    // Write-once 51 MB output: non-temporal so features stay resident.
    __builtin_nontemporal_store(m, out + (size_t)r * PER_ROI + lidx);
}

extern "C" void kernel_launch(void* const* d_in, const int* in_sizes, int n_in,
                              void* d_out, int out_size, void* d_ws, size_t ws_size,
                              hipStream_t stream) {
    (void)n_in; (void)out_size; (void)d_ws; (void)ws_size;
    const float* feat = (const float*)d_in[0];
    const float* rois = (const float*)d_in[1];
    float* out        = (float*)d_out;
    const int R = in_sizes[1] / 5;             // 1024 ROIs
    dim3 grid(R * BLOCKS_PER_ROI);             // 50176 blocks
    dim3 block(256);                           // 8 wave32s per block
    hipLaunchKernelGGL(_RoIPooling_gfx1250_kernel, grid, block, 0, stream,
                       feat, rois, out);
}